// QuadrangleAttention_21045339750859
// MI455X (gfx1250) — compile-verified
//
#include <hip/hip_runtime.h>
#include <hip/hip_bf16.h>
#include <math.h>

// ---------------------------------------------------------------------------
// QuadrangleAttention for MI455X (gfx1250)
//  - fp32 WMMA (V_WMMA_F32_16X16X4_F32) for all matmuls (exact fp32 accum)
//  - async global->LDS copies (GLOBAL_LOAD_ASYNC_TO_LDS_B128) + double buffer
// b=2, n=3136 (56x56), dm=512, nh=8, dh=64, windows 7x7 (64 windows)
// ---------------------------------------------------------------------------

typedef float v2f __attribute__((ext_vector_type(2)));
typedef float v8f __attribute__((ext_vector_type(8)));

static __device__ __forceinline__ v8f v8zero() {
    v8f z;
#pragma unroll
    for (int i = 0; i < 8; ++i) z[i] = 0.0f;
    return z;
}

static __device__ __forceinline__ v8f wmma_f32(v2f a, v2f b, v8f c) {
    // D = A(16x4,f32) * B(4x16,f32) + C(16x16,f32)
    return __builtin_amdgcn_wmma_f32_16x16x4_f32(
        /*neg_a=*/false, a, /*neg_b=*/false, b,
        /*c_mod=*/(short)0, c, /*reuse_a=*/false, /*reuse_b=*/false);
}

// --- async global->LDS (ASYNCcnt) with sync fallback ------------------------
#if defined(__has_builtin)
#if __has_builtin(__builtin_amdgcn_global_load_async_to_lds_b128) && \
    __has_builtin(__builtin_amdgcn_s_wait_asynccnt)
#define HAVE_ASYNC_LDS 1
#endif
#endif
#ifndef HAVE_ASYNC_LDS
#define HAVE_ASYNC_LDS 0
#endif

#if HAVE_ASYNC_LDS
// builtin expects: (int4 addrspace(1)* src, int4 addrspace(3)* dst, imm off, imm cpol)
typedef int v4i_t __attribute__((vector_size(16)));
typedef __attribute__((address_space(1))) v4i_t g_v4i;
typedef __attribute__((address_space(3))) v4i_t l_v4i;
#endif

static __device__ __forceinline__ void async_cp_b128(const float* src, float* lds_dst) {
#if HAVE_ASYNC_LDS
    __builtin_amdgcn_global_load_async_to_lds_b128(
        (g_v4i*)src, (l_v4i*)lds_dst, /*offset=*/0, /*cpol=*/0);
#else
    float4 v = *(const float4*)src;
    *(float4*)lds_dst = v;
#endif
}

#if HAVE_ASYNC_LDS
#define WAIT_ASYNC(n) __builtin_amdgcn_s_wait_asynccnt(n)
#else
#define WAIT_ASYNC(n) ((void)0)
#endif

#define B_SZ     2
#define HH       56
#define WW_      56
#define NTOK     3136      // 56*56
#define DM       512
#define NH       8
#define DH       64
#define NWIN     64        // 8*8 windows of 7x7
#define WS_      49
#define MROWS    6272      // B_SZ * NTOK

// ---------------------------------------------------------------------------
// Generic WMMA GEMM:  C[M,N] = A[M,K] * W[N,K]^T (+ bias[n])
// Block: 128 threads (4 waves), tile 64(M) x 64(N), K chunks of 16,
// double-buffered async staging into LDS.
// M % 64 == 0, N % 64 == 0, K % 16 == 0 required.
// ---------------------------------------------------------------------------
__global__ __launch_bounds__(128)
void qa_gemm_wmma(const float* __restrict__ A, const float* __restrict__ W,
                  const float* __restrict__ bias, float* __restrict__ C,
                  int M, int N, int K) {
    // row stride 20 floats: 16B-aligned rows (80B), banks 20n mod 64 distinct
    __shared__ float As[2][64][20];
    __shared__ float Ws[2][64][20];

    const int tid  = threadIdx.x;
    const int lane = tid & 31;
    const int wid  = tid >> 5;
    const int wm   = wid >> 1;          // 0..1  (wave row)
    const int wn   = wid & 1;           // 0..1  (wave col)
    const int half = lane >> 4;         // 0..1
    const int l16  = lane & 15;

    const int bm = blockIdx.y * 64;
    const int bn = blockIdx.x * 64;

    v8f acc[2][2];
#pragma unroll
    for (int i = 0; i < 2; ++i)
#pragma unroll
        for (int j = 0; j < 2; ++j) acc[i][j] = v8zero();

    // stage one K-chunk (64x16 of A and of W) into LDS buffer `buf`
    auto stage = [&](int kc, int buf) {
        const int kbase = kc << 4;
#pragma unroll
        for (int i = 0; i < 2; ++i) {
            int idx = tid + (i << 7);           // 0..255 float4 slots
            int r   = idx >> 2;
            int c4  = (idx & 3) << 2;
            async_cp_b128(A + (size_t)(bm + r) * K + kbase + c4, &As[buf][r][c4]);
            async_cp_b128(W + (size_t)(bn + r) * K + kbase + c4, &Ws[buf][r][c4]);
        }
    };

    const int kChunks = K >> 4;
    stage(0, 0);
    for (int kc = 0; kc < kChunks; ++kc) {
        const int cur = kc & 1;
        if (kc + 1 < kChunks) {
            stage(kc + 1, cur ^ 1);     // prefetch next chunk (4 async ops)
            WAIT_ASYNC(4);              // current chunk's copies done
        } else {
            WAIT_ASYNC(0);
        }
        __syncthreads();

#pragma unroll
        for (int step = 0; step < 4; ++step) {
            const int ko = (step << 2) + (half << 1);
            v2f afr[2], bfr[2];
#pragma unroll
            for (int i = 0; i < 2; ++i)
                afr[i] = *(v2f*)&As[cur][wm * 32 + i * 16 + l16][ko];
#pragma unroll
            for (int j = 0; j < 2; ++j)
                bfr[j] = *(v2f*)&Ws[cur][wn * 32 + j * 16 + l16][ko];
#pragma unroll
            for (int i = 0; i < 2; ++i)
#pragma unroll
                for (int j = 0; j < 2; ++j)
                    acc[i][j] = wmma_f32(afr[i], bfr[j], acc[i][j]);
        }
        __syncthreads();
    }

    // epilogue: C layout — VGPR v holds row (v + 8*half), col = l16
#pragma unroll
    for (int i = 0; i < 2; ++i)
#pragma unroll
        for (int j = 0; j < 2; ++j) {
            const int col = bn + wn * 32 + j * 16 + l16;
            float bv = (bias != nullptr) ? bias[col] : 0.0f;
#pragma unroll
            for (int v = 0; v < 8; ++v) {
                const int row = bm + wm * 32 + i * 16 + v + (half << 3);
                C[(size_t)row * N + col] = acc[i][j][v] + bv;
            }
        }
}

// ---------------------------------------------------------------------------
// Window mean-pool: pooled[b,win,c] = mean_{7x7}(x)
// grid = 128 (b*win), block = 256
// ---------------------------------------------------------------------------
__global__ __launch_bounds__(256)
void qa_pool(const float* __restrict__ x, float* __restrict__ pooled) {
    const int bw  = blockIdx.x;
    const int b   = bw >> 6;
    const int win = bw & 63;
    const int wy  = win >> 3;
    const int wx  = win & 7;
    for (int c = threadIdx.x; c < DM; c += 256) {
        float s = 0.0f;
#pragma unroll
        for (int iy = 0; iy < 7; ++iy) {
            const size_t base =
                ((size_t)b * NTOK + (size_t)(wy * 7 + iy) * WW_ + wx * 7) * DM + c;
#pragma unroll
            for (int ix = 0; ix < 7; ++ix) s += x[base + (size_t)ix * DM];
        }
        pooled[(size_t)bw * DM + c] = s * (1.0f / 49.0f);
    }
}

// ---------------------------------------------------------------------------
// Transform head + grid + regularizer. grid_dim = 128 (b*win), block = 512
// ---------------------------------------------------------------------------
static __device__ __forceinline__ void m3mul(const float* A, const float* B, float* C) {
#pragma unroll
    for (int i = 0; i < 3; ++i)
#pragma unroll
        for (int j = 0; j < 3; ++j)
            C[i * 3 + j] = A[i * 3 + 0] * B[0 * 3 + j] +
                           A[i * 3 + 1] * B[1 * 3 + j] +
                           A[i * 3 + 2] * B[2 * 3 + j];
}

__global__ __launch_bounds__(512)
void qa_grid(const float* __restrict__ pooled, const float* __restrict__ w_qpm,
             const float* __restrict__ b_qpm, float* __restrict__ grid,
             float* __restrict__ reg_out) {
    __shared__ float sp[DM];
    __shared__ float tl[72];
    __shared__ float Tm[8][9];
    __shared__ float red[512];

    const int tid = threadIdx.x;
    const int bw  = blockIdx.x;
    const int b   = bw >> 6;
    const int win = bw & 63;
    const int wy  = win >> 3;
    const int wx  = win & 7;

    // silu(pooled)
    {
        float v = pooled[(size_t)bw * DM + tid];
        sp[tid] = v / (1.0f + expf(-v));
    }
    __syncthreads();

    // t = silu(pooled) @ w_qpm^T + b_qpm
    if (tid < 72) {
        const float* wr = w_qpm + (size_t)tid * DM;
        float s = b_qpm[tid];
        for (int c = 0; c < DM; ++c) s += sp[c] * wr[c];
        tl[tid] = s;
    }
    __syncthreads();

    // 3x3 transform chain per head
    if (tid < 8) {
        const int h = tid;
        const float sx  = tl[0 * 8 + h], sy  = tl[1 * 8 + h];
        const float shx = tl[2 * 8 + h], shy = tl[3 * 8 + h];
        const float rot = tl[4 * 8 + h];
        const float tx  = tl[5 * 8 + h], ty  = tl[6 * 8 + h];
        const float px  = tl[7 * 8 + h], py  = tl[8 * 8 + h];
        const float cr = cosf(rot), sr = sinf(rot);
        float M1[9] = {sx + 1.f, 0.f, 0.f,  0.f, sy + 1.f, 0.f,  0.f, 0.f, 1.f};
        float M2[9] = {1.f, shx, 0.f,  shy, 1.f, 0.f,  0.f, 0.f, 1.f};
        float M3[9] = {cr, -sr, 0.f,  sr, cr, 0.f,  0.f, 0.f, 1.f};
        float M4[9] = {1.f, 0.f, tx,  0.f, 1.f, ty,  0.f, 0.f, 1.f};
        float M5[9] = {1.f, 0.f, 0.f,  0.f, 1.f, 0.f,  px, py, 1.f};
        float t1[9], t2[9];
        m3mul(M1, M2, t1);
        m3mul(t1, M3, t2);
        m3mul(t2, M4, t1);
        m3mul(t1, M5, t2);
#pragma unroll
        for (int i = 0; i < 9; ++i) Tm[h][i] = t2[i];
    }
    __syncthreads();

    float regLocal = 0.0f;
    if (tid < NH * WS_) {
        const int h = tid / WS_;
        const int s = tid - h * WS_;
        const int iy = s / 7, ix = s - iy * 7;
        const float cy0 = (float)iy - 3.5f;
        const float cx0 = (float)ix - 3.5f;
        const float* T = Tm[h];
        const float h0 = T[0] * cy0 + T[1] * cx0 + T[2];
        const float h1 = T[3] * cy0 + T[4] * cx0 + T[5];
        const float h2 = T[6] * cy0 + T[7] * cx0 + T[8];
        const float inv = 1.0f / (h2 + 1e-6f);
        const float yv = h0 * inv, xv = h1 * inv;
        const float cy = (float)(wy * 7) + 3.5f;
        const float cx = (float)(wx * 7) + 3.5f;
        const float ny = (yv + cy) * (2.0f / 56.0f) - 1.0f;
        const float nx = (xv + cx) * (2.0f / 56.0f) - 1.0f;
        const size_t gi = ((size_t)(b * NH + h) * NTOK + (size_t)win * WS_ + s) * 2;
        grid[gi + 0] = nx;   // grid = normed[..., ::-1]
        grid[gi + 1] = ny;
        float rx = fmaxf(fabsf(nx) - 1.0f, 0.0f);
        float ry = fmaxf(fabsf(ny) - 1.0f, 0.0f);
        regLocal = rx * rx + ry * ry;
    }
    red[tid] = regLocal;
    __syncthreads();
    for (int st = 256; st > 0; st >>= 1) {
        if (tid < st) red[tid] += red[tid + st];
        __syncthreads();
    }
    if (tid == 0) atomicAdd(reg_out, red[0]);   // LAMBDA_REG = 1.0
}

// ---------------------------------------------------------------------------
// Bilinear sampling of K and V at the deformed grid points.
// One thread per (b,h,key); 50176 threads total. grid = 196, block = 256.
// ---------------------------------------------------------------------------
__global__ __launch_bounds__(256)
void qa_sample(const float* __restrict__ qkv, const float* __restrict__ grid,
               float* __restrict__ ks, float* __restrict__ vs) {
    const int idx  = blockIdx.x * 256 + threadIdx.x;   // < 50176
    const int bh   = idx / NTOK;
    const int kidx = idx - bh * NTOK;
    const int b = bh >> 3, h = bh & 7;

    const float gx = grid[(size_t)idx * 2 + 0];
    const float gy = grid[(size_t)idx * 2 + 1];
    const float fx = (gx + 1.0f) * 0.5f * 55.0f;
    const float fy = (gy + 1.0f) * 0.5f * 55.0f;
    const float fx0 = floorf(fx), fy0 = floorf(fy);
    const int x0 = (int)fx0, y0 = (int)fy0;
    const int x1 = x0 + 1,  y1 = y0 + 1;
    const float wx1 = fx - fx0, wx0 = 1.0f - wx1;
    const float wy1 = fy - fy0, wy0 = 1.0f - wy1;

    int   yy[4] = {y0, y0, y1, y1};
    int   xx[4] = {x0, x1, x0, x1};
    float ww[4] = {wy0 * wx0, wy0 * wx1, wy1 * wx0, wy1 * wx1};
    const float* kbase[4];
    const float* vbase[4];
#pragma unroll
    for (int i = 0; i < 4; ++i) {
        const bool valid = (yy[i] >= 0) && (yy[i] < HH) && (xx[i] >= 0) && (xx[i] < WW_);
        if (!valid) ww[i] = 0.0f;
        const int yc = min(max(yy[i], 0), HH - 1);
        const int xc = min(max(xx[i], 0), WW_ - 1);
        const size_t row = ((size_t)b * NTOK + (size_t)yc * WW_ + xc) * (3 * DM);
        kbase[i] = qkv + row + DM + h * DH;          // k block
        vbase[i] = qkv + row + 2 * DM + h * DH;      // v block
    }

    float* kd = ks + ((size_t)bh * NTOK + kidx) * DH;
    float* vd = vs + ((size_t)bh * NTOK + kidx) * DH;
#pragma unroll
    for (int c4 = 0; c4 < DH / 4; ++c4) {
        float4 ak = make_float4(0.f, 0.f, 0.f, 0.f);
        float4 av = make_float4(0.f, 0.f, 0.f, 0.f);
#pragma unroll
        for (int i = 0; i < 4; ++i) {
            const float w = ww[i];
            float4 kk = *(const float4*)(kbase[i] + c4 * 4);
            float4 vv = *(const float4*)(vbase[i] + c4 * 4);
            ak.x = fmaf(w, kk.x, ak.x); ak.y = fmaf(w, kk.y, ak.y);
            ak.z = fmaf(w, kk.z, ak.z); ak.w = fmaf(w, kk.w, ak.w);
            av.x = fmaf(w, vv.x, av.x); av.y = fmaf(w, vv.y, av.y);
            av.z = fmaf(w, vv.z, av.z); av.w = fmaf(w, vv.w, av.w);
        }
        *(float4*)(kd + c4 * 4) = ak;
        *(float4*)(vd + c4 * 4) = av;
    }
}

// ---------------------------------------------------------------------------
// Flash attention, fp32 WMMA, double-buffered async K/V tiles.
// One wave (32 threads) per 16-query tile. grid = (196, 16).
// 1/sqrt(dh) applied to scores (not Q) so Q can be DMA'd untouched.
// ---------------------------------------------------------------------------
__global__ __launch_bounds__(32)
void qa_flash(const float* __restrict__ qkv, const float* __restrict__ ks,
              const float* __restrict__ vs, float* __restrict__ attn_out) {
    // row stride 68 floats: 272B rows (16B aligned), banks 4n mod 64 distinct
    __shared__ float Qs[16][68];
    __shared__ float Ks[2][16][68];
    __shared__ float Vs[2][16][68];
    __shared__ float Ps[16][18];

    const int qt   = blockIdx.x;
    const int bh   = blockIdx.y;
    const int b    = bh >> 3, h = bh & 7;
    const int lane = threadIdx.x;
    const int half = lane >> 4;
    const int l16  = lane & 15;

    // async-load Q tile (16x64) : 8 b128 ops per lane
    const float* qbase = qkv + ((size_t)(b * NTOK + qt * 16)) * (3 * DM) + h * DH;
#pragma unroll
    for (int i = 0; i < 8; ++i) {
        const int fi = lane + (i << 5);          // 0..255 float4 slots
        const int r  = fi >> 4;
        const int c4 = (fi & 15) << 2;
        async_cp_b128(qbase + (size_t)r * (3 * DM) + c4, &Qs[r][c4]);
    }

    auto load_kv = [&](int j, int buf) {
        const size_t kb = ((size_t)bh * NTOK + (size_t)j * 16) * DH;
#pragma unroll
        for (int i = 0; i < 8; ++i) {
            const int fi = lane + (i << 5);
            const int r  = fi >> 4;
            const int c4 = (fi & 15) << 2;
            async_cp_b128(ks + kb + (size_t)r * DH + c4, &Ks[buf][r][c4]);
            async_cp_b128(vs + kb + (size_t)r * DH + c4, &Vs[buf][r][c4]);
        }
    };
    load_kv(0, 0);

    float m_i[8], l_i[8];
#pragma unroll
    for (int v = 0; v < 8; ++v) { m_i[v] = -1e30f; l_i[v] = 0.0f; }
    v8f o[4];
#pragma unroll
    for (int f = 0; f < 4; ++f) o[f] = v8zero();

    for (int j = 0; j < NTOK / 16; ++j) {
        const int cur = j & 1;
        if (j + 1 < NTOK / 16) {
            load_kv(j + 1, cur ^ 1);    // prefetch next tile (16 async ops)
            WAIT_ASYNC(16);             // Q + current tile copies done
        } else {
            WAIT_ASYNC(0);
        }
        __syncthreads();

        // S(16x16) = Q(16x64) * K^T  via 16 fp32 WMMAs
        v8f s = v8zero();
#pragma unroll
        for (int step = 0; step < 16; ++step) {
            const int ko = (step << 2) + (half << 1);
            v2f a  = *(v2f*)&Qs[l16][ko];
            v2f bb = *(v2f*)&Ks[cur][l16][ko];   // B = K^T: B[k][n] = Ks[n][k]
            s = wmma_f32(a, bb, s);
        }

        // online softmax; C-frag row v maps to m = v + 8*half, col = l16
        float scl[8];
#pragma unroll
        for (int v = 0; v < 8; ++v) {
            float sv = s[v] * 0.125f;            // / sqrt(dh)
            float rm = sv;
#pragma unroll
            for (int d = 1; d < 16; d <<= 1) rm = fmaxf(rm, __shfl_xor(rm, d));
            const float mnew = fmaxf(m_i[v], rm);
            const float sc   = expf(m_i[v] - mnew);
            const float p    = expf(sv - mnew);
            float rs = p;
#pragma unroll
            for (int d = 1; d < 16; d <<= 1) rs += __shfl_xor(rs, d);
            l_i[v] = l_i[v] * sc + rs;
            m_i[v] = mnew;
            scl[v] = sc;
            Ps[v + (half << 3)][l16] = p;
        }
#pragma unroll
        for (int f = 0; f < 4; ++f)
#pragma unroll
            for (int v = 0; v < 8; ++v) o[f][v] *= scl[v];
        __syncthreads();

        // O += P(16x16) * V(16x64)
#pragma unroll
        for (int f = 0; f < 4; ++f) {
#pragma unroll
            for (int step = 0; step < 4; ++step) {
                const int ko = (step << 2) + (half << 1);
                v2f a = *(v2f*)&Ps[l16][ko];
                v2f bb;
                bb[0] = Vs[cur][ko + 0][f * 16 + l16];
                bb[1] = Vs[cur][ko + 1][f * 16 + l16];
                o[f] = wmma_f32(a, bb, o[f]);
            }
        }
        __syncthreads();
    }

    // normalize and write out in (b, n, h*64 + d) layout
#pragma unroll
    for (int v = 0; v < 8; ++v) {
        const float inv = 1.0f / l_i[v];
        const int row = qt * 16 + v + (half << 3);
#pragma unroll
        for (int f = 0; f < 4; ++f) {
            const int col = h * DH + f * 16 + l16;
            attn_out[((size_t)b * NTOK + row) * DM + col] = o[f][v] * inv;
        }
    }
}

// ---------------------------------------------------------------------------
// Host-side launch
// ---------------------------------------------------------------------------
extern "C" void kernel_launch(void* const* d_in, const int* in_sizes, int n_in,
                              void* d_out, int out_size, void* d_ws, size_t ws_size,
                              hipStream_t stream) {
    const float* x      = (const float*)d_in[0];
    const float* w_qkv  = (const float*)d_in[1];
    const float* w_out  = (const float*)d_in[2];
    const float* b_out  = (const float*)d_in[3];
    const float* w_qpm  = (const float*)d_in[4];
    const float* b_qpm  = (const float*)d_in[5];
    float* out = (float*)d_out;
    float* ws  = (float*)d_ws;

    // workspace layout (floats); total ~19.4M floats (~78 MB)
    size_t off = 0;
    float* qkv    = ws + off; off += (size_t)MROWS * (3 * DM);   // 9,633,792
    float* pooled = ws + off; off += (size_t)B_SZ * NWIN * DM;   //    65,536
    float* grid   = ws + off; off += (size_t)B_SZ * NH * NTOK * 2;
    float* ksb    = ws + off; off += (size_t)B_SZ * NH * NTOK * DH;
    float* vsb    = ws + off; off += (size_t)B_SZ * NH * NTOK * DH;
    float* attn   = ws + off; off += (size_t)MROWS * DM;
    (void)off; (void)ws_size; (void)in_sizes; (void)n_in; (void)out_size;

    float* reg_out = out + (size_t)MROWS * DM;   // last element of d_out

    // 1) QKV projection: (6272x512) @ (1536x512)^T
    qa_gemm_wmma<<<dim3(3 * DM / 64, MROWS / 64), 128, 0, stream>>>(
        x, w_qkv, nullptr, qkv, MROWS, 3 * DM, DM);

    // 2) window pooling
    qa_pool<<<B_SZ * NWIN, 256, 0, stream>>>(x, pooled);

    // 3) transform head + grid + reg
    (void)hipMemsetAsync(reg_out, 0, sizeof(float), stream);
    qa_grid<<<B_SZ * NWIN, 512, 0, stream>>>(pooled, w_qpm, b_qpm, grid, reg_out);

    // 4) bilinear sampling of k/v
    qa_sample<<<(B_SZ * NH * NTOK) / 256, 256, 0, stream>>>(qkv, grid, ksb, vsb);

    // 5) flash attention (fp32 WMMA, online softmax, async double-buffer)
    qa_flash<<<dim3(NTOK / 16, B_SZ * NH), 32, 0, stream>>>(qkv, ksb, vsb, attn);

    // 6) output projection + bias -> d_out
    qa_gemm_wmma<<<dim3(DM / 64, MROWS / 64), 128, 0, stream>>>(
        attn, w_out, b_out, out, MROWS, DM, DM);
}